// REN_7937099563009
// MI455X (gfx1250) — compile-verified
//
#include <hip/hip_runtime.h>
#include <hip/hip_bf16.h>

#define N_DIM 2048
#define L_DIM 4096
#define M_DIM 512
#define P_DIM 512

typedef float v2f __attribute__((ext_vector_type(2)));
typedef float v8f __attribute__((ext_vector_type(8)));
typedef unsigned int u32x4 __attribute__((ext_vector_type(4)));
typedef int i32x4 __attribute__((ext_vector_type(4)));
typedef int i32x8 __attribute__((ext_vector_type(8)));

__device__ __forceinline__ float wave_reduce(float v) {
#pragma unroll
  for (int m = 16; m >= 1; m >>= 1) v += __shfl_xor(v, m, 32);
  return v;  // all 32 lanes hold the sum
}

__device__ __forceinline__ float dot_row(const float* __restrict__ row,
                                         const float* __restrict__ vec,
                                         int n, int lane) {
  const float4* r4 = (const float4*)row;
  const float4* v4 = (const float4*)vec;
  int n4 = n >> 2;
  float acc = 0.f;
  for (int j = lane; j < n4; j += 32) {
    __builtin_prefetch((const void*)(r4 + j + 32), 0, 3);  // global_prefetch_b8
    float4 a = r4[j], b = v4[j];
    acc = fmaf(a.x, b.x, acc);
    acc = fmaf(a.y, b.y, acc);
    acc = fmaf(a.z, b.z, acc);
    acc = fmaf(a.w, b.w, acc);
  }
  return acc;
}

__device__ __forceinline__ float decay_of(const int* __restrict__ t) {
  return __powf(0.95f, (float)(*t));
}

// Issue a TDM load of a 2D fp32 tile [rows x cols] from a row-major matrix
// (leading dim ldm elements) at `gsrc` into LDS at byte-offset `lds_addr`.
// D# layout per CDNA5 ISA.
__device__ __forceinline__ void tdm_load_tile_f32(unsigned lds_addr,
                                                  const float* gsrc,
                                                  unsigned rows, unsigned cols,
                                                  unsigned tensor_w, unsigned tensor_h,
                                                  unsigned ldm) {
  unsigned long long ga = (unsigned long long)gsrc;
  u32x4 g0;
  g0[0] = 1u;                                        // count=1, user descriptor
  g0[1] = lds_addr;                                  // lds_addr [63:32]
  g0[2] = (unsigned)(ga & 0xFFFFFFFFu);              // global_addr[31:0]
  g0[3] = (unsigned)((ga >> 32) & 0x1FFFFFFu) | (2u << 30);  // addr[56:32] | type=2
  i32x8 g1;
  g1[0] = (int)(2u << 16);                           // data_size = 4B (code 2)
  g1[1] = (int)((tensor_w & 0xFFFFu) << 16);         // tensor_dim0[15:0]
  g1[2] = (int)(((tensor_w >> 16) & 0xFFFFu) | ((tensor_h & 0xFFFFu) << 16));
  g1[3] = (int)(((tensor_h >> 16) & 0xFFFFu) | ((cols & 0xFFFFu) << 16));  // tile_dim0
  g1[4] = (int)(rows & 0xFFFFu);                     // tile_dim1 (tile_dim2 = 0)
  g1[5] = (int)ldm;                                  // tensor_dim0_stride[31:0]
  g1[6] = 0;                                         // stride[47:32] | dim1_stride[15:0]
  g1[7] = 0;                                         // dim1_stride[47:16]
  i32x4 gz = {0, 0, 0, 0};
#if __clang_major__ >= 23
  i32x8 gz8 = {0, 0, 0, 0, 0, 0, 0, 0};
  __builtin_amdgcn_tensor_load_to_lds(g0, g1, gz, gz, gz8, 0);
#else
  __builtin_amdgcn_tensor_load_to_lds(g0, g1, gz, gz, 0);
#endif
}

// a[i] = C1[i,:]@x + D12[i,:]@u + decay*bv[i]
__global__ void k_gemv_a(const float* __restrict__ C1, const float* __restrict__ D12,
                         const float* __restrict__ x, const float* __restrict__ u,
                         const float* __restrict__ bv, const int* __restrict__ t,
                         float* __restrict__ a_out) {
  int lane = threadIdx.x & 31;
  int row = blockIdx.x * 8 + (threadIdx.x >> 5);
  float acc = dot_row(C1 + (size_t)row * N_DIM, x, N_DIM, lane)
            + dot_row(D12 + (size_t)row * M_DIM, u, M_DIM, lane);
  acc = wave_reduce(acc);
  if (lane == 0) a_out[row] = acc + decay_of(t) * bv[row];
}

// Blocked triangular tanh forward substitution. One workgroup (32 waves).
// eps[i] = tanh((a[i] + D11[i,:i]@eps[:i]) / Lambda[i])
__global__ void __launch_bounds__(1024)
k_scan(const float* __restrict__ a, const float* __restrict__ D11,
       const float* __restrict__ Lambda, float* __restrict__ eps_out) {
  __shared__ float eps_s[L_DIM];
  __shared__ float part_s[32];
  __shared__ float tile_s[32 * 33];  // 32x32 diagonal tile, padded
  int tid = threadIdx.x;
  int wave = tid >> 5, lane = tid & 31;
  for (int blk = 0; blk < L_DIM / 32; ++blk) {
    int base = blk * 32;
    int row = base + wave;
    const float* rp = D11 + (size_t)row * L_DIM;
    // panel partial: dot of row's first `base` entries with known eps
    float acc = 0.f;
    for (int j = lane; j < base; j += 32) acc = fmaf(rp[j], eps_s[j], acc);
    acc = wave_reduce(acc);
    if (lane == 0) part_s[wave] = acc + a[row];
    // stage 32x32 diagonal tile into LDS
    tile_s[wave * 33 + lane] = rp[base + lane];
    __syncthreads();
    if (wave == 0) {
      float eps_reg = 0.f;                 // lane j will hold eps[base+j]
      float lam = Lambda[base + lane];
      for (int i = 0; i < 32; ++i) {
        float term = (lane < i) ? tile_s[i * 33 + lane] * eps_reg : 0.f;
        term = wave_reduce(term);
        float li = __shfl(lam, i, 32);
        float val = tanhf((part_s[i] + term) / li);
        if (lane == i) eps_reg = val;
      }
      eps_s[base + lane] = eps_reg;
    }
    __syncthreads();
  }
  for (int j = tid; j < L_DIM; j += 1024) eps_out[j] = eps_s[j];
}

// E_x[i] = F[i,:]@x + B1[i,:]@eps + B2[i,:]@u + decay*bx[i]
__global__ void k_gemv_Ex(const float* __restrict__ F, const float* __restrict__ B1,
                          const float* __restrict__ B2, const float* __restrict__ x,
                          const float* __restrict__ eps, const float* __restrict__ u,
                          const float* __restrict__ bx, const int* __restrict__ t,
                          float* __restrict__ out) {
  int lane = threadIdx.x & 31;
  int row = blockIdx.x * 8 + (threadIdx.x >> 5);
  float acc = dot_row(F + (size_t)row * N_DIM, x, N_DIM, lane)
            + dot_row(B1 + (size_t)row * L_DIM, eps, L_DIM, lane)
            + dot_row(B2 + (size_t)row * M_DIM, u, M_DIM, lane);
  acc = wave_reduce(acc);
  if (lane == 0) out[row] = acc + decay_of(t) * bx[row];
}

// y[p] = C2[p,:]@x + D21[p,:]@eps + D22[p,:]@u + decay*bu[p]
__global__ void k_gemv_y(const float* __restrict__ C2, const float* __restrict__ D21,
                         const float* __restrict__ D22, const float* __restrict__ x,
                         const float* __restrict__ eps, const float* __restrict__ u,
                         const float* __restrict__ bu, const int* __restrict__ t,
                         float* __restrict__ out) {
  int lane = threadIdx.x & 31;
  int row = blockIdx.x * 8 + (threadIdx.x >> 5);
  float acc = dot_row(C2 + (size_t)row * N_DIM, x, N_DIM, lane)
            + dot_row(D21 + (size_t)row * L_DIM, eps, L_DIM, lane)
            + dot_row(D22 + (size_t)row * M_DIM, u, M_DIM, lane);
  acc = wave_reduce(acc);
  if (lane == 0) out[row] = acc + decay_of(t) * bu[row];
}

// Delta = E - I
__global__ void k_make_delta(const float* __restrict__ E, float* __restrict__ D) {
  int idx = blockIdx.x * blockDim.x + threadIdx.x;
  int row = idx >> 11, col = idx & (N_DIM - 1);
  D[idx] = E[idx] - ((row == col) ? 1.0f : 0.0f);
}

// out[i] = b[i] + s * (P[i,:] @ v)
__global__ void k_apply(const float* __restrict__ b, const float* __restrict__ v,
                        const float* __restrict__ P, float* __restrict__ out, float s) {
  int lane = threadIdx.x & 31;
  int row = blockIdx.x * 8 + (threadIdx.x >> 5);
  float acc = wave_reduce(dot_row(P + (size_t)row * N_DIM, v, N_DIM, lane));
  if (lane == 0) out[row] = b[row] + s * acc;
}

// C = A @ B, square n=2048, fp32 via V_WMMA_F32_16X16X4_F32.
// WG = 128 threads = 4 waves; wave w computes the 16x16 tile at
// (blockIdx.y*16, blockIdx.x*64 + w*16).
// B panel (64 k-rows x 64 cols, 16KB) is staged into LDS by the Tensor Data
// Mover (wave 0 issues the DMA, waits TENSORcnt==0, workgroup barrier
// publishes). A-frags come straight from global (shared by all 4 waves -> L0).
// A-frag (16x4): lane l -> row = l&15, k-pair = (l>>4)*2 (VGPR0=K0/K2, VGPR1=K1/K3).
// B-frag (4x16): dual layout, lane l -> col = l&15, k-pair = (l>>4)*2.
__global__ void __launch_bounds__(128)
k_gemm_sq(const float* __restrict__ A, const float* __restrict__ B,
          float* __restrict__ C) {
  __shared__ float Bs[64 * 64];
  const int n = N_DIM;
  int tid = threadIdx.x;
  int wave = tid >> 5, lane = tid & 31;
  int r = lane & 15, half = lane >> 4;
  int row0 = blockIdx.y << 4;
  int colWG = blockIdx.x << 6;

  const float* aptr = A + (size_t)(row0 + r) * n + (half << 1);
  unsigned lds_base = (unsigned)(unsigned long long)(void*)Bs;  // LDS byte offset
  const float* bpanel = B + colWG;

  // per-lane LDS read offsets into the 64x64 tile
  int bcol = (wave << 4) + r;
  int brow = half << 1;

  v8f acc = {};
  for (int kc = 0; kc < n; kc += 64) {
    __syncthreads();  // previous tile fully consumed
    if (wave == 0) {
      tdm_load_tile_f32(lds_base, bpanel + (size_t)kc * n,
                        /*rows=*/64, /*cols=*/64,
                        /*tensor_w=*/n, /*tensor_h=*/n, /*ldm=*/n);
      __builtin_amdgcn_s_wait_tensorcnt((short)0);
    }
    __syncthreads();  // tile visible to all waves
#pragma unroll
    for (int kk = 0; kk < 64; kk += 4) {
      v2f af = *(const v2f*)aptr;            // A[row0+r, k+2h .. +1]
      v2f bf;
      bf.x = Bs[(kk + brow) * 64 + bcol];    // B[k+2h,   col]
      bf.y = Bs[(kk + brow + 1) * 64 + bcol];
      acc = __builtin_amdgcn_wmma_f32_16x16x4_f32(
          false, af, false, bf, (short)0, acc, false, false);
      aptr += 4;
    }
  }
  // D 16x16 f32 layout: VGPR v, lane l -> row = v + 8*(l>>4), col = l&15
  float* cptr = C + (size_t)(row0 + (half << 3)) * n + (colWG + (wave << 4) + r);
#pragma unroll
  for (int i = 0; i < 8; ++i) cptr[(size_t)i * n] = acc[i];
}

extern "C" void kernel_launch(void* const* d_in, const int* in_sizes, int n_in,
                              void* d_out, int out_size, void* d_ws, size_t ws_size,
                              hipStream_t stream) {
  const float* x   = (const float*)d_in[0];
  const float* u   = (const float*)d_in[1];
  const float* C1  = (const float*)d_in[2];
  const float* D11 = (const float*)d_in[3];
  const float* D12 = (const float*)d_in[4];
  const float* Lam = (const float*)d_in[5];
  const float* F   = (const float*)d_in[6];
  const float* B1  = (const float*)d_in[7];
  const float* B2  = (const float*)d_in[8];
  const float* E   = (const float*)d_in[9];
  const float* C2  = (const float*)d_in[10];
  const float* D21 = (const float*)d_in[11];
  const float* D22 = (const float*)d_in[12];
  const float* bv  = (const float*)d_in[13];
  const float* bx  = (const float*)d_in[14];
  const float* bu  = (const float*)d_in[15];
  const int*   t   = (const int*)d_in[16];
  float* out = (float*)d_out;            // [y(512) | x_(2048)]
  float* ws  = (float*)d_ws;

  float* a_ws = ws;                                    // 4096
  float* eps  = ws + 4096;                             // 4096
  float* exv  = ws + 8192;                             // 2048
  float* v1   = ws + 10240;                            // 2048
  float* v0   = ws + 12288;                            // 2048
  float* Dl   = ws + 16384;                            // 2048^2
  float* Pa   = Dl + (size_t)N_DIM * N_DIM;            // 2048^2
  float* Pb   = Pa + (size_t)N_DIM * N_DIM;            // 2048^2

  dim3 gg(N_DIM / 64, N_DIM / 16);

  k_make_delta<<<(N_DIM * N_DIM) / 256, 256, 0, stream>>>(E, Dl);
  k_gemv_a<<<L_DIM / 8, 256, 0, stream>>>(C1, D12, x, u, bv, t, a_ws);
  k_scan<<<1, 1024, 0, stream>>>(a_ws, D11, Lam, eps);
  k_gemv_Ex<<<N_DIM / 8, 256, 0, stream>>>(F, B1, B2, x, eps, u, bx, t, exv);
  k_gemv_y<<<P_DIM / 8, 256, 0, stream>>>(C2, D21, D22, x, eps, u, bu, t, out);

  // inv(I+D) = (I-D)(I+D^2)(I+D^4)...(I+D^128), error O(D^256)
  k_apply<<<N_DIM / 8, 256, 0, stream>>>(exv, exv, Dl, v1, -1.0f);  // (I-D)b
  k_gemm_sq<<<gg, 128, 0, stream>>>(Dl, Dl, Pa);                    // D^2
  k_apply<<<N_DIM / 8, 256, 0, stream>>>(v1, v1, Pa, v0, 1.0f);
  k_gemm_sq<<<gg, 128, 0, stream>>>(Pa, Pa, Pb);                    // D^4
  k_apply<<<N_DIM / 8, 256, 0, stream>>>(v0, v0, Pb, v1, 1.0f);
  k_gemm_sq<<<gg, 128, 0, stream>>>(Pb, Pb, Pa);                    // D^8
  k_apply<<<N_DIM / 8, 256, 0, stream>>>(v1, v1, Pa, v0, 1.0f);
  k_gemm_sq<<<gg, 128, 0, stream>>>(Pa, Pa, Pb);                    // D^16
  k_apply<<<N_DIM / 8, 256, 0, stream>>>(v0, v0, Pb, v1, 1.0f);
  k_gemm_sq<<<gg, 128, 0, stream>>>(Pb, Pb, Pa);                    // D^32
  k_apply<<<N_DIM / 8, 256, 0, stream>>>(v1, v1, Pa, v0, 1.0f);
  k_gemm_sq<<<gg, 128, 0, stream>>>(Pa, Pa, Pb);                    // D^64
  k_apply<<<N_DIM / 8, 256, 0, stream>>>(v0, v0, Pb, v1, 1.0f);
  k_gemm_sq<<<gg, 128, 0, stream>>>(Pb, Pb, Pa);                    // D^128
  k_apply<<<N_DIM / 8, 256, 0, stream>>>(v1, v1, Pa, out + P_DIM, 1.0f);  // x_

  (void)in_sizes; (void)n_in; (void)out_size; (void)ws_size;
}